// GNN_46918222741705
// MI455X (gfx1250) — compile-verified
//
#include <hip/hip_runtime.h>

#define N_NODES 100000
#define N_EDGES 1200000
#define NB      128
#define H       64
#define NTILES  (N_NODES / 16)   /* 6250 exactly */
#define GN_EPS  1e-5f

typedef float v2f __attribute__((ext_vector_type(2)));
typedef float v8f __attribute__((ext_vector_type(8)));

// ---------------------------------------------------------------- utilities
__device__ __forceinline__ int f2ord(float f) {
    int i = __float_as_int(f);
    return i >= 0 ? i : (i ^ 0x7fffffff);
}
__device__ __forceinline__ float ord2f(int i) {
    return __int_as_float(i >= 0 ? i : (i ^ 0x7fffffff));
}

__global__ void fill_f32(float* __restrict__ p, float v, int n) {
    int i = blockIdx.x * blockDim.x + threadIdx.x;
    if (i < n) p[i] = v;
}
__global__ void fill_i32(int* __restrict__ p, int v, int n) {
    int i = blockIdx.x * blockDim.x + threadIdx.x;
    if (i < n) p[i] = v;
}

// ------------------------------------------------- GIN aggregation (scatter)
// thread = (edge, feature-quad); 16 lanes cover one edge's 64 features (256B)
__global__ void gin_scatter(const float* __restrict__ xin,
                            const int* __restrict__ src,
                            const int* __restrict__ dst,
                            float* __restrict__ agg) {
    long long tid = (long long)blockIdx.x * blockDim.x + threadIdx.x;
    if (tid >= (long long)N_EDGES * 16) return;
    int e = (int)(tid >> 4);
    int f = ((int)tid & 15) << 2;
    int s = src[e], d = dst[e];
    const float4 v = *(const float4*)(xin + (long long)s * H + f);
    float* a = agg + (long long)d * H + f;
    atomicAdd(a + 0, v.x);
    atomicAdd(a + 1, v.y);
    atomicAdd(a + 2, v.z);
    atomicAdd(a + 3, v.w);
}

// ------------------------------------------- fused GIN MLP on the WMMA pipe
// out = relu((x+agg) @ W1 + b1) @ W2 + b2, f32 WMMA 16x16x4, out may alias agg.
// Weights are pre-swizzled into LDS in B-fragment order: one ds_load_b64 per
// fragment, no cross-register shuffles before v_wmma.
__global__ __launch_bounds__(128) void gin_mlp_wmma(
    const float* __restrict__ xin, const float* __restrict__ agg,
    const float* __restrict__ w1, const float* __restrict__ b1,
    const float* __restrict__ w2, const float* __restrict__ b2,
    float* __restrict__ out) {
    __shared__ float lWs1[2 * 2048];   // 2048 float2 pairs, B-frag order
    __shared__ float lWs2[2 * 2048];
    __shared__ float lB1[H];
    __shared__ float lB2[H];
    __shared__ float lH[4][16 * H];

    const int tid = threadIdx.x;
    // swizzle preload: pair p = (kc*4 + nt)*32 + lane
    //   col = nt*16 + (lane&15), k0 = kc*4 + ((lane>>4)<<1)
    //   pair = { W[k0*H+col], W[(k0+1)*H+col] }
    for (int p = tid; p < 2048; p += 128) {
        int lane = p & 31;
        int q    = p >> 5;        // kc*4 + nt
        int nt   = q & 3;
        int kc   = q >> 2;
        int col  = nt * 16 + (lane & 15);
        int k0   = kc * 4 + ((lane >> 4) << 1);
        lWs1[p * 2 + 0] = w1[k0 * H + col];
        lWs1[p * 2 + 1] = w1[(k0 + 1) * H + col];
        lWs2[p * 2 + 0] = w2[k0 * H + col];
        lWs2[p * 2 + 1] = w2[(k0 + 1) * H + col];
    }
    if (tid < H) { lB1[tid] = b1[tid]; lB2[tid] = b2[tid]; }
    __syncthreads();

    const int wid  = tid >> 5;
    const int lane = tid & 31;
    const int tile = blockIdx.x * 4 + wid;
    const bool valid = tile < NTILES;
    const int row0 = (valid ? tile : 0) * 16;

    const int m     = lane & 15;          // M row (A) / N col (B,C)
    const int koff  = (lane >> 4) << 1;   // K sub-offset per half-wave
    const int mbase = (lane >> 4) << 3;   // C-layout M base

    // ---- A = x + agg in 16x16x4 A-layout (16 k-chunks of float2)
    v2f a[16];
    {
        const float* xr = xin + (long long)(row0 + m) * H + koff;
        const float* ar = agg + (long long)(row0 + m) * H + koff;
#pragma unroll
        for (int kc = 0; kc < 16; ++kc) {
            v2f xv = *(const v2f*)(xr + kc * 4);
            v2f av = *(const v2f*)(ar + kc * 4);
            a[kc].x = xv.x + av.x;
            a[kc].y = xv.y + av.y;
        }
    }

    // ---- GEMM1: acc = A @ W1 + b1
    v8f acc[4];
#pragma unroll
    for (int nt = 0; nt < 4; ++nt) {
        float bv = lB1[nt * 16 + m];
#pragma unroll
        for (int i = 0; i < 8; ++i) acc[nt][i] = bv;
    }
#pragma unroll
    for (int kc = 0; kc < 16; ++kc) {
#pragma unroll
        for (int nt = 0; nt < 4; ++nt) {
            v2f bf = *(const v2f*)(&lWs1[(((kc << 2) + nt) << 6) + lane * 2]);
            acc[nt] = __builtin_amdgcn_wmma_f32_16x16x4_f32(
                false, a[kc], false, bf, (short)0, acc[nt], false, false);
        }
    }

    // ---- ReLU, round-trip through LDS to convert C-layout -> A-layout
    float* hb = &lH[wid][0];
#pragma unroll
    for (int nt = 0; nt < 4; ++nt)
#pragma unroll
        for (int r = 0; r < 8; ++r) {
            float v = acc[nt][r];
            hb[(mbase + r) * H + nt * 16 + m] = v > 0.f ? v : 0.f;
        }
    // same-wave DS ops are in-order: safe to read back without a barrier
#pragma unroll
    for (int kc = 0; kc < 16; ++kc) {
        a[kc] = *(const v2f*)(&hb[m * H + kc * 4 + koff]);
    }

    // ---- GEMM2: acc = H @ W2 + b2
#pragma unroll
    for (int nt = 0; nt < 4; ++nt) {
        float bv = lB2[nt * 16 + m];
#pragma unroll
        for (int i = 0; i < 8; ++i) acc[nt][i] = bv;
    }
#pragma unroll
    for (int kc = 0; kc < 16; ++kc) {
#pragma unroll
        for (int nt = 0; nt < 4; ++nt) {
            v2f bf = *(const v2f*)(&lWs2[(((kc << 2) + nt) << 6) + lane * 2]);
            acc[nt] = __builtin_amdgcn_wmma_f32_16x16x4_f32(
                false, a[kc], false, bf, (short)0, acc[nt], false, false);
        }
    }

    if (valid) {
#pragma unroll
        for (int nt = 0; nt < 4; ++nt)
#pragma unroll
            for (int r = 0; r < 8; ++r)
                out[(long long)(row0 + mbase + r) * H + nt * 16 + m] = acc[nt][r];
    }
}

// ----------------------------------------------------------- GraphNorm
__global__ void gn_stats1(const float* __restrict__ x, const int* __restrict__ batch,
                          float* __restrict__ sum, float* __restrict__ cnt) {
    int tid = blockIdx.x * blockDim.x + threadIdx.x;
    if (tid >= N_NODES * 16) return;
    int n = tid >> 4, f = (tid & 15) << 2;
    int g = batch[n];
    float4 v = *(const float4*)(x + (long long)n * H + f);
    float* s = sum + g * H + f;
    atomicAdd(s + 0, v.x);
    atomicAdd(s + 1, v.y);
    atomicAdd(s + 2, v.z);
    atomicAdd(s + 3, v.w);
    if ((tid & 15) == 0) atomicAdd(cnt + g, 1.0f);
}

__global__ void gn_finalize_mean(float* __restrict__ mean, const float* __restrict__ cnt) {
    int i = blockIdx.x * blockDim.x + threadIdx.x;
    if (i >= NB * H) return;
    float c = cnt[i >> 6];
    mean[i] /= (c < 1.f ? 1.f : c);
}

__global__ void gn_stats2(const float* __restrict__ x, const int* __restrict__ batch,
                          const float* __restrict__ mean, const float* __restrict__ ms,
                          float* __restrict__ var) {
    int tid = blockIdx.x * blockDim.x + threadIdx.x;
    if (tid >= N_NODES * 16) return;
    int n = tid >> 4, f = (tid & 15) << 2;
    int g = batch[n];
    float4 v = *(const float4*)(x + (long long)n * H + f);
    float* vv = var + g * H + f;
    float o0 = v.x - ms[f + 0] * mean[g * H + f + 0];
    float o1 = v.y - ms[f + 1] * mean[g * H + f + 1];
    float o2 = v.z - ms[f + 2] * mean[g * H + f + 2];
    float o3 = v.w - ms[f + 3] * mean[g * H + f + 3];
    atomicAdd(vv + 0, o0 * o0);
    atomicAdd(vv + 1, o1 * o1);
    atomicAdd(vv + 2, o2 * o2);
    atomicAdd(vv + 3, o3 * o3);
}

__global__ void gn_finalize_var(float* __restrict__ var, const float* __restrict__ cnt) {
    int i = blockIdx.x * blockDim.x + threadIdx.x;
    if (i >= NB * H) return;
    float c = cnt[i >> 6];
    c = c < 1.f ? 1.f : c;
    var[i] = rsqrtf(var[i] / c + GN_EPS);   // store 1/sqrt(var+eps)
}

__global__ void gn_norm_relu(const float* __restrict__ x, const int* __restrict__ batch,
                             const float* __restrict__ mean, const float* __restrict__ rstd,
                             const float* __restrict__ ms, const float* __restrict__ w,
                             const float* __restrict__ b, float* __restrict__ out) {
    int tid = blockIdx.x * blockDim.x + threadIdx.x;
    if (tid >= N_NODES * 16) return;
    int n = tid >> 4, f = (tid & 15) << 2;
    int g = batch[n];
    float4 v = *(const float4*)(x + (long long)n * H + f);
    float4 o;
    float* vp = (float*)&v;
    float* op = (float*)&o;
#pragma unroll
    for (int j = 0; j < 4; ++j) {
        float t = (vp[j] - ms[f + j] * mean[g * H + f + j]) * (rstd[g * H + f + j] * w[f + j]) + b[f + j];
        op[j] = t > 0.f ? t : 0.f;
    }
    *(float4*)(out + (long long)n * H + f) = o;
}

// ------------------------------------------------------------- pooling path
__global__ void gate_kernel(const float* __restrict__ h1, const float* __restrict__ h2,
                            const float* __restrict__ h3, const int* __restrict__ batch,
                            const float* __restrict__ gw, const float* __restrict__ gb,
                            float* __restrict__ hsum, float* __restrict__ gate,
                            int* __restrict__ gmax) {
    int n = blockIdx.x * blockDim.x + threadIdx.x;
    if (n >= N_NODES) return;
    float s = gb[0];
    long long base = (long long)n * H;
    for (int f = 0; f < H; f += 4) {
        float4 a = *(const float4*)(h1 + base + f);
        float4 b = *(const float4*)(h2 + base + f);
        float4 c = *(const float4*)(h3 + base + f);
        float4 w = *(const float4*)(gw + f);
        float4 hv;
        hv.x = a.x + b.x + c.x;
        hv.y = a.y + b.y + c.y;
        hv.z = a.z + b.z + c.z;
        hv.w = a.w + b.w + c.w;
        *(float4*)(hsum + base + f) = hv;
        s += hv.x * w.x + hv.y * w.y + hv.z * w.z + hv.w * w.w;
    }
    gate[n] = s;
    atomicMax(gmax + batch[n], f2ord(s));
}

__global__ void pool_kernel(const float* __restrict__ hsum, const float* __restrict__ gate,
                            const int* __restrict__ batch, const int* __restrict__ gmax,
                            float* __restrict__ attnum, float* __restrict__ meansum,
                            int* __restrict__ maxp, float* __restrict__ den) {
    int tid = blockIdx.x * blockDim.x + threadIdx.x;
    if (tid >= N_NODES * 16) return;
    int n = tid >> 4, f = (tid & 15) << 2;
    int g = batch[n];
    float ex = __expf(gate[n] - ord2f(gmax[g]));
    float4 hv = *(const float4*)(hsum + (long long)n * H + f);
    float* an = attnum + g * H + f;
    float* mn = meansum + g * H + f;
    int* mp = maxp + g * H + f;
    atomicAdd(an + 0, ex * hv.x);
    atomicAdd(an + 1, ex * hv.y);
    atomicAdd(an + 2, ex * hv.z);
    atomicAdd(an + 3, ex * hv.w);
    atomicAdd(mn + 0, hv.x);
    atomicAdd(mn + 1, hv.y);
    atomicAdd(mn + 2, hv.z);
    atomicAdd(mn + 3, hv.w);
    atomicMax(mp + 0, f2ord(hv.x));
    atomicMax(mp + 1, f2ord(hv.y));
    atomicMax(mp + 2, f2ord(hv.z));
    atomicMax(mp + 3, f2ord(hv.w));
    if ((tid & 15) == 0) atomicAdd(den + g, ex);
}

// ------------------------------------------------------------------- head
__global__ __launch_bounds__(64) void head_kernel(
    const float* __restrict__ attnum, const float* __restrict__ meansum,
    const int* __restrict__ maxp, const float* __restrict__ den,
    const float* __restrict__ cnt, const float* __restrict__ u,
    const float* __restrict__ hw1, const float* __restrict__ hb1,
    const float* __restrict__ hw2, const float* __restrict__ hb2,
    float* __restrict__ out) {
    int g = blockIdx.x, j = threadIdx.x;
    __shared__ float red[H];
    float c = cnt[g];
    c = c < 1.f ? 1.f : c;
    float dn = den[g];
    float acc = hb1[j];
    for (int k = 0; k < H; ++k) acc += (attnum[g * H + k] / dn) * hw1[k * H + j];
    for (int k = 0; k < H; ++k) acc += (meansum[g * H + k] / c) * hw1[(H + k) * H + j];
    for (int k = 0; k < H; ++k) acc += ord2f(maxp[g * H + k]) * hw1[(2 * H + k) * H + j];
    for (int k = 0; k < 3; ++k) acc += u[g * 3 + k] * hw1[(3 * H + k) * H + j];
    acc = acc > 0.f ? acc : 0.f;
    red[j] = acc * hw2[j];
    __syncthreads();
    if (j == 0) {
        float s = hb2[0];
        for (int k = 0; k < H; ++k) s += red[k];
        out[g] = s;
    }
}

// ------------------------------------------------------------------ driver
extern "C" void kernel_launch(void* const* d_in, const int* in_sizes, int n_in,
                              void* d_out, int out_size, void* d_ws, size_t ws_size,
                              hipStream_t stream) {
    (void)in_sizes; (void)n_in; (void)out_size; (void)ws_size;
    const float* x     = (const float*)d_in[0];
    const int*   src   = (const int*)d_in[1];
    const int*   dst   = src + N_EDGES;
    const int*   batch = (const int*)d_in[2];
    const float* u     = (const float*)d_in[3];
    const float* gate_w = (const float*)d_in[25];
    const float* gate_b = (const float*)d_in[26];
    const float* hw1    = (const float*)d_in[27];
    const float* hb1    = (const float*)d_in[28];
    const float* hw2    = (const float*)d_in[29];
    const float* hb2    = (const float*)d_in[30];

    const size_t NH = (size_t)N_NODES * H;
    const size_t BH = (size_t)NB * H;
    float* agg     = (float*)d_ws;
    float* h1      = agg + NH;
    float* h2      = h1 + NH;
    float* h3      = h2 + NH;
    float* mean    = h3 + NH;
    float* rstd    = mean + BH;
    float* cnt     = rstd + BH;
    float* gate    = cnt + NB;
    float* den     = gate + N_NODES;
    float* attnum  = den + NB;
    float* meansum = attnum + BH;
    int*   maxp    = (int*)(meansum + BH);
    int*   gmax    = maxp + BH;

    const float* lay_in[3] = {x, h1, h2};
    float*       lay_out[3] = {h1, h2, h3};

    for (int l = 0; l < 3; ++l) {
        const float* w1a = (const float*)d_in[4 + 7 * l + 0];
        const float* b1a = (const float*)d_in[4 + 7 * l + 1];
        const float* w1b = (const float*)d_in[4 + 7 * l + 2];
        const float* b1b = (const float*)d_in[4 + 7 * l + 3];
        const float* gnw = (const float*)d_in[4 + 7 * l + 4];
        const float* gnb = (const float*)d_in[4 + 7 * l + 5];
        const float* gnm = (const float*)d_in[4 + 7 * l + 6];

        fill_f32<<<(int)((NH + 255) / 256), 256, 0, stream>>>(agg, 0.f, (int)NH);
        fill_f32<<<(int)((BH + 255) / 256), 256, 0, stream>>>(mean, 0.f, (int)BH);
        fill_f32<<<(int)((BH + 255) / 256), 256, 0, stream>>>(rstd, 0.f, (int)BH);
        fill_f32<<<1, 256, 0, stream>>>(cnt, 0.f, NB);

        gin_scatter<<<N_EDGES * 16 / 256, 256, 0, stream>>>(lay_in[l], src, dst, agg);
        gin_mlp_wmma<<<(NTILES + 3) / 4, 128, 0, stream>>>(lay_in[l], agg, w1a, b1a, w1b, b1b, agg);

        gn_stats1<<<N_NODES * 16 / 256, 256, 0, stream>>>(agg, batch, mean, cnt);
        gn_finalize_mean<<<(int)((BH + 255) / 256), 256, 0, stream>>>(mean, cnt);
        gn_stats2<<<N_NODES * 16 / 256, 256, 0, stream>>>(agg, batch, mean, gnm, rstd);
        gn_finalize_var<<<(int)((BH + 255) / 256), 256, 0, stream>>>(rstd, cnt);
        gn_norm_relu<<<N_NODES * 16 / 256, 256, 0, stream>>>(agg, batch, mean, rstd, gnm, gnw, gnb, lay_out[l]);
    }

    fill_i32<<<1, 256, 0, stream>>>(gmax, (int)0x80000000, NB);
    fill_f32<<<1, 256, 0, stream>>>(den, 0.f, NB);
    fill_f32<<<(int)((BH + 255) / 256), 256, 0, stream>>>(attnum, 0.f, (int)BH);
    fill_f32<<<(int)((BH + 255) / 256), 256, 0, stream>>>(meansum, 0.f, (int)BH);
    fill_i32<<<(int)((BH + 255) / 256), 256, 0, stream>>>(maxp, (int)0x80000000, (int)BH);

    gate_kernel<<<(N_NODES + 255) / 256, 256, 0, stream>>>(h1, h2, h3, batch, gate_w, gate_b,
                                                           agg /*hsum*/, gate, gmax);
    pool_kernel<<<N_NODES * 16 / 256, 256, 0, stream>>>(agg, gate, batch, gmax,
                                                        attnum, meansum, maxp, den);
    head_kernel<<<NB, 64, 0, stream>>>(attnum, meansum, maxp, den, cnt, u,
                                       hw1, hb1, hw2, hb2, (float*)d_out);
}